// Discriminator_74156905332812
// MI455X (gfx1250) — compile-verified
//
#include <hip/hip_runtime.h>
#include <math.h>

// Problem constants (from the reference)
#define IN_DIM 256
#define HID    64
#define BIAS_C 0.0001f

// Padded LDS row stride for transposed W: must be >= IN_DIM and == 4 (mod 64)
// so B-fragment ds_load_b64 reads hit all 64 banks exactly once.
#define WT_STRIDE 260

typedef float v2f __attribute__((ext_vector_type(2)));
typedef float v8f __attribute__((ext_vector_type(8)));

// ---------------------------------------------------------------------------
// Kernel 1: t[i] = relu(emb[i,:] @ W_emb + b_emb) @ wc     (wc = 0.5*(w1+w2))
// One wave -> 16 nodes x 64 hidden, via V_WMMA_F32_16X16X4_F32.
// W_emb is staged TRANSPOSED in LDS: sWt[col][k], row stride 260 floats, so a
// B fragment {W[kk][col], W[kk+1][col]} is one conflict-free ds_load_b64.
// ---------------------------------------------------------------------------
__global__ __launch_bounds__(256) void node_score_kernel(
    const float* __restrict__ emb,     // [N, 256]
    const float* __restrict__ W,       // [256, 64]
    const float* __restrict__ b_emb,   // [64]
    const float* __restrict__ W_edge,  // [128]
    float* __restrict__ t,             // [N] out
    int n_nodes)
{
    __shared__ float sWt[HID * WT_STRIDE];   // 66,560 B (< 320 KB WGP limit)

    const int tid = threadIdx.x;

    // Cooperative transposed fill: global reads coalesced, LDS writes scattered
    // (one-time cost per block).
    #pragma unroll
    for (int i = tid; i < IN_DIM * HID; i += 256) {
        const int k = i >> 6;     // row of W   (0..255)
        const int c = i & 63;     // col of W   (0..63)
        sWt[c * WT_STRIDE + k] = W[i];
    }
    __syncthreads();

    const int wave = tid >> 5;
    const int lane = tid & 31;
    const int nodeBase = blockIdx.x * 128 + wave * 16;
    if (nodeBase >= n_nodes) return;      // uniform per wave: EXEC stays all-1s

    const int half = lane >> 4;           // 0: lanes 0-15, 1: lanes 16-31
    const int l16  = lane & 15;

    // Per-lane bias / combined edge weight for the 4 hidden-column chunks.
    float bias0 = b_emb[l16 +  0], bias1 = b_emb[l16 + 16];
    float bias2 = b_emb[l16 + 32], bias3 = b_emb[l16 + 48];
    float wc0 = 0.5f * (W_edge[l16 +  0] + W_edge[HID + l16 +  0]);
    float wc1 = 0.5f * (W_edge[l16 + 16] + W_edge[HID + l16 + 16]);
    float wc2 = 0.5f * (W_edge[l16 + 32] + W_edge[HID + l16 + 32]);
    float wc3 = 0.5f * (W_edge[l16 + 48] + W_edge[HID + l16 + 48]);

    const float* arow = emb + (size_t)(nodeBase + l16) * IN_DIM;

    // Per-lane base pointers into the transposed LDS tile for each chunk.
    const float* bcol0 = sWt + (l16 +  0) * WT_STRIDE;
    const float* bcol1 = sWt + (l16 + 16) * WT_STRIDE;
    const float* bcol2 = sWt + (l16 + 32) * WT_STRIDE;
    const float* bcol3 = sWt + (l16 + 48) * WT_STRIDE;

    v8f acc0 = {}, acc1 = {}, acc2 = {}, acc3 = {};

    // A 16x4 f32 frag: lanes 0-15 -> K = k0,k0+1 ; lanes 16-31 -> K = k0+2,k0+3
    // B 4x16 f32 frag: same K split across lane halves, N = column.
    #pragma unroll 4
    for (int k0 = 0; k0 < IN_DIM; k0 += 4) {
        const int kk = k0 + (half << 1);
        v2f a  = *(const v2f*)(arow  + kk);   // global b64, coalesced
        v2f b0 = *(const v2f*)(bcol0 + kk);   // ds_load_b64, conflict-free
        v2f b1 = *(const v2f*)(bcol1 + kk);
        v2f b2 = *(const v2f*)(bcol2 + kk);
        v2f b3 = *(const v2f*)(bcol3 + kk);

        acc0 = __builtin_amdgcn_wmma_f32_16x16x4_f32(false, a, false, b0, (short)0, acc0, false, false);
        acc1 = __builtin_amdgcn_wmma_f32_16x16x4_f32(false, a, false, b1, (short)0, acc1, false, false);
        acc2 = __builtin_amdgcn_wmma_f32_16x16x4_f32(false, a, false, b2, (short)0, acc2, false, false);
        acc3 = __builtin_amdgcn_wmma_f32_16x16x4_f32(false, a, false, b3, (short)0, acc3, false, false);
    }

    // Epilogue: +bias, relu, *wc, reduce each row of the 16x64 tile.
    // C/D layout: VGPR r <-> row M=r (lanes 0-15) / M=r+8 (lanes 16-31), N=l16.
    float rsum[8];
    #pragma unroll
    for (int r = 0; r < 8; ++r) {
        float s = fmaxf(acc0[r] + bias0, 0.f) * wc0
                + fmaxf(acc1[r] + bias1, 0.f) * wc1
                + fmaxf(acc2[r] + bias2, 0.f) * wc2
                + fmaxf(acc3[r] + bias3, 0.f) * wc3;
        s += __shfl_xor(s, 1, 32);
        s += __shfl_xor(s, 2, 32);
        s += __shfl_xor(s, 4, 32);
        s += __shfl_xor(s, 8, 32);   // lanes 0 / 16 now hold row sums
        rsum[r] = s;
    }
    if (l16 == 0) {
        const int m = half << 3;     // 0 or 8
        #pragma unroll
        for (int r = 0; r < 8; ++r)
            t[nodeBase + m + r] = rsum[r];
    }
}

// ---------------------------------------------------------------------------
// Kernel 2: streaming edge pass.
// out[e] = sigmoid(log(eps) - log1p(-eps) + t[src] + t[dst] + b_edge)
// ---------------------------------------------------------------------------
__global__ __launch_bounds__(256) void edge_gate_kernel(
    const int*   __restrict__ edges,   // [2, E]
    const float* __restrict__ noise,   // [E]
    const float* __restrict__ b_edge,  // [1]
    const float* __restrict__ t,       // [N]
    float* __restrict__ out,           // [E]
    int E)
{
    const int e = blockIdx.x * 256 + threadIdx.x;
    if (e >= E) return;

    const int   si = edges[e];
    const int   di = edges[E + e];
    const float s  = t[si] + t[di] + b_edge[0];

    const float u   = noise[e];
    const float eps = (2.0f * BIAS_C - 1.0f) * u + (1.0f - BIAS_C);
    const float gate = __logf(eps) - log1pf(-eps) + s;   // TEMPERATURE == 1
    out[e] = 1.0f / (1.0f + __expf(-gate));
}

// ---------------------------------------------------------------------------
// Launcher
// ---------------------------------------------------------------------------
extern "C" void kernel_launch(void* const* d_in, const int* in_sizes, int n_in,
                              void* d_out, int out_size, void* d_ws, size_t ws_size,
                              hipStream_t stream)
{
    const float* emb    = (const float*)d_in[0];
    const int*   edges  = (const int*)  d_in[1];
    const float* noise  = (const float*)d_in[2];
    const float* W_emb  = (const float*)d_in[3];
    const float* b_emb  = (const float*)d_in[4];
    const float* W_edge = (const float*)d_in[5];
    const float* b_edge = (const float*)d_in[6];
    float*       out    = (float*)d_out;

    const int n_nodes = in_sizes[0] / IN_DIM;   // 50000
    const int E       = in_sizes[2];            // 1600000

    float* t = (float*)d_ws;                    // n_nodes floats of scratch

    // Phase 1: per-node scalar scores via f32 WMMA (128 nodes per block).
    const int blocks1 = (n_nodes + 127) / 128;
    node_score_kernel<<<blocks1, 256, 0, stream>>>(emb, W_emb, b_emb, W_edge,
                                                   t, n_nodes);

    // Phase 2: streaming edge gate.
    const int blocks2 = (E + 255) / 256;
    edge_gate_kernel<<<blocks2, 256, 0, stream>>>(edges, noise, b_edge, t,
                                                  out, E);
}